// SpatialShiftConvBlock_71923522339050
// MI455X (gfx1250) — compile-verified
//
#include <hip/hip_runtime.h>
#include <hip/hip_bf16.h>

typedef __attribute__((ext_vector_type(2))) float v2f;
typedef __attribute__((ext_vector_type(8))) float v8f;

#define M_ROWS    10752   // B*T*N = 8*64*21
#define K_DIM     128     // C_in
#define F_DIM     256     // F
#define N_SPATIAL 21
#define BN_EPS    1e-3f

// ---------------------------------------------------------------------------
// Kernel 0: zero the BN accumulator region of the workspace (ws[0..511]).
// ---------------------------------------------------------------------------
__global__ void ssc_zero_kernel(float* __restrict__ ws) {
    int i = blockIdx.x * blockDim.x + threadIdx.x;
    if (i < 512) ws[i] = 0.0f;
}

// ---------------------------------------------------------------------------
// Kernel 1: y = spatial_shift(x) @ W + b  via V_WMMA_F32_16X16X4_F32.
//
// Block = 128 threads = 4 waves. Block tile = 64(M) x 64(F):
//   wave w owns columns (blockIdx.y*4 + w)*16, and iterates 4 M-subtiles,
//   holding 4 v8f accumulators so each B-fragment is reused by 4 WMMAs
//   (4x less W traffic out of L2 vs a 16-row block tile).
// The shifted A-tile (64 rows x 128 ch) is staged once in LDS (row stride
// 132 floats: bank = (132*row + k) % 64 = (4*row + k) % 64, conflict-free
// for the 16-row column-strided fragment reads).
// ---------------------------------------------------------------------------
__global__ __launch_bounds__(128) void ssc_gemm_kernel(
    const float* __restrict__ x,      // [M_ROWS, K_DIM] viewed as [BT, N, C]
    const float* __restrict__ W,      // [K_DIM, F_DIM]
    const float* __restrict__ bias,   // [F_DIM]
    float* __restrict__ y)            // [M_ROWS, F_DIM]
{
    __shared__ float tile[64][132];   // 33 KB of the 320 KB WGP LDS

    const int tid     = threadIdx.x;
    const int lane    = tid & 31;
    const int wave    = tid >> 5;
    const int rowBase = blockIdx.x * 64;
    const int colBase = (blockIdx.y * 4 + wave) * 16;

    // ---- stage shifted x tile: thread t owns channel c = t for all 64 rows
    {
        const int c = tid;                 // 0..127 == channel
        const int s = c % N_SPATIAL;       // circular shift for this channel
        #pragma unroll 4
        for (int i = 0; i < 64; ++i) {
            int gr   = rowBase + i;
            int bt   = gr / N_SPATIAL;
            int n    = gr - bt * N_SPATIAL;
            int srcn = n - s; if (srcn < 0) srcn += N_SPATIAL;
            tile[i][c] = x[(bt * N_SPATIAL + srcn) * K_DIM + c];
        }
    }
    __syncthreads();

    const int h    = lane >> 4;            // lane-half selects K pair / M half
    const int ml   = lane & 15;            // A row (M) / B,C column (N)
    const int colB = colBase + ml;         // global output channel

    v8f acc[4];
    #pragma unroll
    for (int mt = 0; mt < 4; ++mt) acc[mt] = (v8f){};

    #pragma unroll 2
    for (int kk = 0; kk < K_DIM; kk += 4) {
        const int k0 = kk + 2 * h;
        // B 4x16 fragment: VGPR v -> K = 2*h + v, col N = ml (shared by 4 WMMAs)
        v2f bf;
        bf.x = W[(size_t)k0 * F_DIM + colB];
        bf.y = W[(size_t)(k0 + 1) * F_DIM + colB];
        // 4 A fragments (one per M-subtile) + 4 chained WMMAs
        #pragma unroll
        for (int mt = 0; mt < 4; ++mt) {
            v2f a;
            a.x = tile[mt * 16 + ml][k0];
            a.y = tile[mt * 16 + ml][k0 + 1];
            acc[mt] = __builtin_amdgcn_wmma_f32_16x16x4_f32(
                /*neg_a=*/false, a, /*neg_b=*/false, bf,
                /*c_mod=*/(short)0, acc[mt], /*reuse_a=*/false, /*reuse_b=*/false);
        }
    }

    // Epilogue: bias add + store pre-BN activations (no atomics here).
    const float bv = bias[colB];
    #pragma unroll
    for (int mt = 0; mt < 4; ++mt) {
        #pragma unroll
        for (int r = 0; r < 8; ++r) {
            float v = acc[mt][r] + bv;     // D row M = mt*16 + r + 8*h
            y[(size_t)(rowBase + mt * 16 + r + 8 * h) * F_DIM + colB] = v;
        }
    }
}

// ---------------------------------------------------------------------------
// Kernel 2: per-channel sum / sum-of-squares over y (L2-resident, 11 MB).
// 84 blocks x 256 threads; thread t owns channel t, register-accumulates
// 128 rows (fully coalesced b32 reads), then ONE atomic per accumulator:
// only 84 same-address atomics per stats slot instead of ~1344.
// ---------------------------------------------------------------------------
__global__ __launch_bounds__(256) void ssc_reduce_kernel(
    const float* __restrict__ y, float* __restrict__ ws)
{
    const int c    = threadIdx.x;          // channel
    const int row0 = blockIdx.x * 128;
    float s1 = 0.0f, s2 = 0.0f;
    #pragma unroll 4
    for (int r = 0; r < 128; ++r) {
        float v = y[(size_t)(row0 + r) * F_DIM + c];
        s1 += v;
        s2 += v * v;
    }
    atomicAdd(&ws[c],         s1);
    atomicAdd(&ws[256 + c],   s2);
}

// ---------------------------------------------------------------------------
// Kernel 3: fold sums into per-channel affine BN coefficients.
//   scale = gamma * rsqrt(var + eps);  shift = beta - mean * scale
// ---------------------------------------------------------------------------
__global__ __launch_bounds__(256) void ssc_stats_kernel(
    const float* __restrict__ gamma, const float* __restrict__ beta,
    float* __restrict__ ws)
{
    const int c = threadIdx.x;             // 0..255
    const float inv_m = 1.0f / (float)M_ROWS;
    const float mean  = ws[c] * inv_m;
    const float var   = ws[256 + c] * inv_m - mean * mean;
    const float scale = gamma[c] * rsqrtf(var + BN_EPS);
    ws[512 + c] = scale;
    ws[768 + c] = beta[c] - mean * scale;
}

// ---------------------------------------------------------------------------
// Kernel 4: in-place BN + ReLU over y, float4-vectorized (F=256 so 4
// consecutive elements are 4 consecutive channels).
// ---------------------------------------------------------------------------
__global__ __launch_bounds__(256) void ssc_apply_kernel(
    float* __restrict__ y, const float* __restrict__ ws)
{
    __shared__ float sc[F_DIM];
    __shared__ float sh[F_DIM];
    sc[threadIdx.x] = ws[512 + threadIdx.x];
    sh[threadIdx.x] = ws[768 + threadIdx.x];
    __syncthreads();

    const size_t total4 = (size_t)M_ROWS * F_DIM / 4;
    const size_t i = (size_t)blockIdx.x * blockDim.x + threadIdx.x;
    if (i < total4) {
        float4* y4 = (float4*)y;
        float4 v = y4[i];
        const int c = (int)((i * 4) & (F_DIM - 1));
        v.x = fmaxf(fmaf(v.x, sc[c    ], sh[c    ]), 0.0f);
        v.y = fmaxf(fmaf(v.y, sc[c + 1], sh[c + 1]), 0.0f);
        v.z = fmaxf(fmaf(v.z, sc[c + 2], sh[c + 2]), 0.0f);
        v.w = fmaxf(fmaf(v.w, sc[c + 3], sh[c + 3]), 0.0f);
        y4[i] = v;
    }
}

// ---------------------------------------------------------------------------
extern "C" void kernel_launch(void* const* d_in, const int* in_sizes, int n_in,
                              void* d_out, int out_size, void* d_ws, size_t ws_size,
                              hipStream_t stream) {
    (void)in_sizes; (void)n_in; (void)out_size; (void)ws_size;
    const float* x     = (const float*)d_in[0];   // [8,64,21,128]
    const float* W     = (const float*)d_in[1];   // [128,256]
    const float* b     = (const float*)d_in[2];   // [256]
    const float* gamma = (const float*)d_in[3];   // [256]
    const float* beta  = (const float*)d_in[4];   // [256]
    float* y  = (float*)d_out;                    // [10752,256]
    float* ws = (float*)d_ws;                     // [0:256)=sum [256:512)=sumsq
                                                  // [512:768)=scale [768:1024)=shift

    ssc_zero_kernel<<<2, 256, 0, stream>>>(ws);

    dim3 grid(M_ROWS / 64, F_DIM / 64);           // (168, 4)
    ssc_gemm_kernel<<<grid, 128, 0, stream>>>(x, W, b, y);

    ssc_reduce_kernel<<<M_ROWS / 128, 256, 0, stream>>>(y, ws);  // 84 blocks

    ssc_stats_kernel<<<1, 256, 0, stream>>>(gamma, beta, ws);

    const int total4 = M_ROWS * F_DIM / 4;        // 688128
    ssc_apply_kernel<<<(total4 + 255) / 256, 256, 0, stream>>>(y, ws);
}